// RoIPool_49847390437672
// MI455X (gfx1250) — compile-verified
//
#include <hip/hip_runtime.h>
#include <stdint.h>

// RoI max pooling (Fast R-CNN semantics), fp32, for MI455X / gfx1250.
// Memory-bound gather+max: no matmul structure -> WMMA not applicable.
// CDNA5 paths: Tensor Data Mover (tensor_load_to_lds, TENSORcnt) for the
// 3D tile stage (16ch x rows x 64), async-LDS fallback, wave32 shuffle reduce.

#define POOL        7
#define SCALE       0.0625f
#define C_TOT       256
#define HF          64
#define WF          64
#define CH_TILE     16                  // channels per block
#define ROWS_MAX    12                  // hend-hstart <= bin_h+2 <= 12 (roi_h<=65)
#define ROW_STRIDE  WF                  // staged row = full 64-float feature row

typedef int  v4i  __attribute__((vector_size(16)));
typedef __attribute__((address_space(1))) v4i glob_v4i;
typedef __attribute__((address_space(3))) v4i lds_v4i;

// TDM descriptor operand types (probe-confirmed shapes)
typedef unsigned int u32x4 __attribute__((ext_vector_type(4)));
typedef int          i32x4 __attribute__((ext_vector_type(4)));
typedef int          i32x8 __attribute__((ext_vector_type(8)));

__device__ __forceinline__ uint32_t lds_byte_offset(const void* p) {
    // generic -> AS3 -> integer gives the workgroup-relative LDS byte address
    return (uint32_t)(size_t)(__attribute__((address_space(3))) const void*)p;
}

__device__ __forceinline__ void async_copy_b128(const float* g, float* l) {
#if __has_builtin(__builtin_amdgcn_global_load_async_to_lds_b128)
    __builtin_amdgcn_global_load_async_to_lds_b128((glob_v4i*)g, (lds_v4i*)l,
                                                   /*imm offset=*/0, /*cpol=*/0);
#else
    asm volatile("global_load_async_to_lds_b128 %0, %1, off"
                 :: "v"((uint32_t)(size_t)l), "v"(g) : "memory");
#endif
}

__device__ __forceinline__ void wait_async0() {
#if __has_builtin(__builtin_amdgcn_s_wait_asynccnt)
    __builtin_amdgcn_s_wait_asynccnt(0);
#else
    asm volatile("s_wait_asynccnt 0" ::: "memory");
#endif
}

__device__ __forceinline__ void wait_tensor0() {
#if __has_builtin(__builtin_amdgcn_s_wait_tensorcnt)
    __builtin_amdgcn_s_wait_tensorcnt(0);
#else
    asm volatile("s_wait_tensorcnt 0" ::: "memory");
#endif
}

__global__ __launch_bounds__(256) void roipool_gfx1250_kernel(
    const float* __restrict__ feat,   // [B, 256, 64, 64]
    const float* __restrict__ rois,   // [N, 5]
    float* __restrict__ out)          // [N, 256, 7, 7]
{
    __shared__ float smem[CH_TILE * ROWS_MAX * ROW_STRIDE];   // 48 KB

    const int n   = blockIdx.x;
    const int ph  = blockIdx.y;
    const int c0  = blockIdx.z * CH_TILE;
    const int tid = threadIdx.x;

    // ---- ROI parameters (uniform across block; matches reference exactly) ----
    const float* r = rois + n * 5;
    const int   bidx = (int)r[0];
    const float x1 = rintf(r[1] * SCALE);   // rintf = RNE, matches jnp.round
    const float y1 = rintf(r[2] * SCALE);
    const float x2 = rintf(r[3] * SCALE);
    const float y2 = rintf(r[4] * SCALE);
    const float roi_w = fmaxf(x2 - x1 + 1.0f, 1.0f);
    const float roi_h = fmaxf(y2 - y1 + 1.0f, 1.0f);
    const float bin_w = roi_w / (float)POOL;
    const float bin_h = roi_h / (float)POOL;

    const float hs_f = fminf(fmaxf(floorf((float)ph * bin_h) + y1, 0.0f), (float)HF);
    const float he_f = fminf(fmaxf(ceilf(((float)ph + 1.0f) * bin_h) + y1, 0.0f), (float)HF);
    const int rh0 = (int)hs_f;
    const int rh1 = (int)he_f;
    int rows = rh1 - rh0;
    if (rows < 0) rows = 0;
    if (rows > ROWS_MAX) rows = ROWS_MAX;   // provably unreachable; LDS safety

    // Tile source: [CH_TILE x rows x 64] at feat[bidx, c0, rh0, 0],
    // strides (4096, 64, 1) floats. Staged contiguously in LDS.
    const float* gbase = feat + ((size_t)bidx * C_TOT + c0) * (size_t)(HF * WF)
                              + (size_t)rh0 * WF;

#if __has_builtin(__builtin_amdgcn_tensor_load_to_lds)
    // ---- One TDM instruction moves the whole 3D tile (wave 0 issues) ----
    if (rows > 0 && tid < 32) {
        const uint64_t ga  = (uint64_t)(size_t)gbase;
        const uint32_t lds = lds_byte_offset(&smem[0]);

        u32x4 g0 = { 1u,                                   // count=1, no gather
                     lds,                                  // lds_addr (bytes)
                     (uint32_t)ga,                         // global_addr[31:0]
                     (uint32_t)((ga >> 32) & 0x01FFFFFFu)  // global_addr[56:32]
                         | (2u << 30) };                   // type=2 ("image")
        i32x8 g1 = { (int)0x00020000,                      // data_size=2 (4B)
                     (int)0x00400000,                      // tensor_dim0=64 (lo16<<16)
                     (int)0x00400000,                      // td0 hi=0 | tensor_dim1=64 lo16
                     (int)0x00400000,                      // td1 hi=0 | tile_dim0=64
                     (int)((uint32_t)rows | (16u << 16)),  // tile_dim1=rows, tile_dim2=16
                     64,                                   // tensor_dim0_stride lo32
                     (int)0x10000000,                      // s0 hi=0 | tensor_dim1_stride lo16=4096
                     0 };                                  // tensor_dim1_stride hi
        i32x4 g2 = { 256, 0, 0, 0 };                       // tensor_dim2=256; dim3/tile3 unused
        i32x4 g3 = { 0, 0, 0, 0 };
#if defined(__clang_major__) && (__clang_major__ >= 23)
        i32x8 g4 = { 0, 0, 0, 0, 0, 0, 0, 0 };
        __builtin_amdgcn_tensor_load_to_lds(g0, g1, g2, g3, g4, 0);
#else
        __builtin_amdgcn_tensor_load_to_lds(g0, g1, g2, g3, 0);
#endif
        wait_tensor0();                 // TDM done writing LDS (wave 0)
    }
    __syncthreads();                    // release the other 7 waves
#else
    // ---- Fallback: async global->LDS b128 staging loop (same LDS layout) ----
    const int chunks_per_ch = rows * (WF / 4);
    const int total_chunks  = CH_TILE * chunks_per_ch;
    for (int i = tid; i < total_chunks; i += 256) {
        const int cl  = i / chunks_per_ch;
        const int rem = i - cl * chunks_per_ch;
        const int rr  = rem >> 4;
        const int ck  = rem & 15;
        const float* g = gbase + cl * (HF * WF) + rr * WF + (ck << 2);
        float*       l = &smem[(cl * rows + rr) * ROW_STRIDE + (ck << 2)];
        async_copy_b128(g, l);
    }
    wait_async0();
    __syncthreads();
#endif

    // ---- Compute: 2 threads per (channel, pw) bin; waves 0..6 fully active ----
    const int ch_stride = rows * ROW_STRIDE;            // contiguous TDM packing
    if (tid < 2 * CH_TILE * POOL) {                     // 224 threads
        const int part = tid & 1;
        const int bin  = tid >> 1;
        const int cl   = bin / POOL;
        const int pw   = bin - cl * POOL;

        const float ws_f = fminf(fmaxf(floorf((float)pw * bin_w) + x1, 0.0f), (float)WF);
        const float we_f = fminf(fmaxf(ceilf(((float)pw + 1.0f) * bin_w) + x1, 0.0f), (float)WF);
        const int ws = (int)ws_f;
        const int we = (int)we_f;
        const bool empty = (he_f <= hs_f) || (we_f <= ws_f);

        float v = -__builtin_huge_valf();
        const float* cbase = &smem[cl * ch_stride];
        for (int rr = part; rr < rows; rr += 2) {
            const float* rowp = cbase + rr * ROW_STRIDE;
            for (int w = ws; w < we; ++w)
                v = fmaxf(v, rowp[w]);
        }
        // pair-combine (lanes 2b / 2b+1 are in the same wave32)
        const float o = __shfl_xor(v, 1, 32);
        v = fmaxf(v, o);

        if (part == 0) {
            out[(((size_t)n * C_TOT + (c0 + cl)) * POOL + ph) * POOL + pw] =
                empty ? 0.0f : v;
        }
    }
}

extern "C" void kernel_launch(void* const* d_in, const int* in_sizes, int n_in,
                              void* d_out, int out_size, void* d_ws, size_t ws_size,
                              hipStream_t stream) {
    const float* feat = (const float*)d_in[0];   // [4,256,64,64] fp32
    const float* rois = (const float*)d_in[1];   // [N,5] fp32
    float* out = (float*)d_out;                  // [N,256,7,7] fp32
    const int N = in_sizes[1] / 5;

    dim3 grid((unsigned)N, POOL, C_TOT / CH_TILE);
    roipool_gfx1250_kernel<<<grid, 256, 0, stream>>>(feat, rois, out);
}